// ClusterMlpDWBN_5471788335923
// MI455X (gfx1250) — compile-verified
//
#include <hip/hip_runtime.h>
#include <hip/hip_bf16.h>
#include <math.h>

// Problem constants (fixed by the reference)
#define BB   8
#define NN   4096
#define N0   16384
#define CIN  64
#define CHID 256
#define COUT 64
#define HH   128
#define WW   128

typedef __attribute__((ext_vector_type(2))) float v2f;
typedef __attribute__((ext_vector_type(8))) float v8f;

__device__ __forceinline__ float gelu_exact(float x) {
    return 0.5f * x * (1.0f + erff(x * 0.70710678118654752f));
}

__device__ __forceinline__ void atomAddF(float* p, float v) {
    unsafeAtomicAdd(p, v);   // native global_atomic_add_f32
}

// ---------------------------------------------------------------------------
// Zero a float region (grid-stride)
// ---------------------------------------------------------------------------
__global__ void zero_f32(float* __restrict__ p, size_t n) {
    size_t i = (size_t)blockIdx.x * blockDim.x + threadIdx.x;
    size_t stride = (size_t)gridDim.x * blockDim.x;
    for (; i < n; i += stride) p[i] = 0.0f;
}

// ---------------------------------------------------------------------------
// FP32 WMMA GEMM + bias:  C[M,N] = A[M,K] * Bw[N,K]^T + bias[N]
// One wave32 per 16x16 output tile, V_WMMA_F32_16X16X4_F32, K/4 steps.
// A-frag (16x4): lane 0-15 -> row M=lane, VGPR0/1 = K{k,k+1}; lanes 16-31 -> K{k+2,k+3}
// B-frag (4x16): lane 0-15 -> col N=lane, same K interleave (mirror of A)
// C/D: VGPR r -> M = r + (lane>=16 ? 8 : 0), N = lane&15
// ---------------------------------------------------------------------------
template <int K>
__global__ void wmma_gemm_bias(const float* __restrict__ A,
                               const float* __restrict__ Bw,
                               const float* __restrict__ bias,
                               float* __restrict__ C,
                               int M, int N) {
    int wave   = (int)((blockIdx.x * blockDim.x + threadIdx.x) >> 5);
    int ntiles = N >> 4;
    int mt = wave / ntiles;
    int nt = wave - mt * ntiles;
    if (mt >= (M >> 4)) return;            // wave-uniform, EXEC stays all-ones

    unsigned lane = threadIdx.x & 31u;
    unsigned half = lane >> 4;             // 0: K even pair, 1: K odd pair (+2)
    unsigned l15  = lane & 15u;

    int row = (mt << 4) + (int)l15;        // A row (M) for this lane
    int col = (nt << 4) + (int)l15;        // B row / output col (N) for this lane

    const float* ap = A  + (size_t)row * K + half * 2u;
    const float* bp = Bw + (size_t)col * K + half * 2u;

    v8f acc = {};
#pragma unroll
    for (int k = 0; k < K; k += 4) {
        v2f a = *(const v2f*)(ap + k);
        v2f b = *(const v2f*)(bp + k);
        acc = __builtin_amdgcn_wmma_f32_16x16x4_f32(
            false, a, false, b, (short)0, acc, false, false);
    }

    float bv = bias[col];
#pragma unroll
    for (int r = 0; r < 8; ++r) {
        int m = (mt << 4) + r + (int)(half * 8u);
        C[(size_t)m * N + col] = acc[r] + bv;
    }
}

// ---------------------------------------------------------------------------
// Per-channel BN statistics: blockDim.x == C (channels fastest dim)
// ---------------------------------------------------------------------------
__global__ void bn_stats(const float* __restrict__ X, float* __restrict__ sum,
                         float* __restrict__ sumsq, int rows) {
    int C = blockDim.x;
    int c = threadIdx.x;
    float s = 0.0f, s2 = 0.0f;
    for (int r = blockIdx.x; r < rows; r += gridDim.x) {
        float v = X[(size_t)r * C + c];
        s += v; s2 += v * v;
    }
    atomAddF(&sum[c], s);
    atomAddF(&sumsq[c], s2);
}

__global__ void bn_finalize(const float* __restrict__ sum, const float* __restrict__ sumsq,
                            const float* __restrict__ g, const float* __restrict__ b,
                            float* __restrict__ scale, float* __restrict__ shift,
                            float inv_n) {
    int c = threadIdx.x;
    float mean = sum[c] * inv_n;
    float var  = sumsq[c] * inv_n - mean * mean;
    float s = g[c] * rsqrtf(var + 1e-5f);
    scale[c] = s;
    shift[c] = b[c] - mean * s;
}

// y = gelu(x*scale[c] + shift[c]), c = i & cmask   (C must be power of 2)
__global__ void bn_gelu_apply(const float* __restrict__ src, float* __restrict__ dst,
                              const float* __restrict__ scale, const float* __restrict__ shift,
                              size_t n, unsigned cmask) {
    size_t i = (size_t)blockIdx.x * blockDim.x + threadIdx.x;
    size_t stride = (size_t)gridDim.x * blockDim.x;
    for (; i < n; i += stride) {
        unsigned c = (unsigned)i & cmask;
        dst[i] = gelu_exact(src[i] * scale[c] + shift[c]);
    }
}

// ---------------------------------------------------------------------------
// Pixel index helper (matches reference _pixel_index: clip, RNE round, clamp)
// ---------------------------------------------------------------------------
__device__ __forceinline__ void point_pixel(const float* __restrict__ loc, int p,
                                            int& px, int& py) {
    float lx = fminf(fmaxf(loc[2 * p],     -1.0f), 1.0f);
    float ly = fminf(fmaxf(loc[2 * p + 1], -1.0f), 1.0f);
    px = (int)rintf(0.5f * (lx + 1.0f) * (float)WW - 0.5f);
    py = (int)rintf(0.5f * (ly + 1.0f) * (float)HH - 0.5f);
    px = min(max(px, 0), WW - 1);
    py = min(max(py, 0), HH - 1);
}

// ---------------------------------------------------------------------------
// token2map scatter: one block (256 thr = channels) per original point
// ---------------------------------------------------------------------------
__global__ void token2map(const float* __restrict__ h, const float* __restrict__ loc,
                          const int* __restrict__ idx_agg, const float* __restrict__ aggw,
                          float* __restrict__ fmap, float* __restrict__ counts,
                          float* __restrict__ wsum) {
    int p = blockIdx.x;                 // [0, B*N0)
    int b = p >> 14;                    // N0 = 16384
    int px, py; point_pixel(loc, p, px, py);
    int pix = (b * HH + py) * WW + px;
    int t   = b * NN + idx_agg[p];
    int c   = threadIdx.x;
    atomAddF(&fmap[(size_t)pix * CHID + c], h[(size_t)t * CHID + c]);
    if (c == 0) atomAddF(&counts[pix], 1.0f);
    if (c == 1) atomAddF(&wsum[t], aggw[p]);
}

// fmap[i] /= counts[pix] + 1e-6   (pix = i / 256)
__global__ void map_norm(float* __restrict__ fmap, const float* __restrict__ counts, size_t n) {
    size_t i = (size_t)blockIdx.x * blockDim.x + threadIdx.x;
    size_t stride = (size_t)gridDim.x * blockDim.x;
    for (; i < n; i += stride)
        fmap[i] = fmap[i] / (counts[i >> 8] + 1e-6f);
}

// ---------------------------------------------------------------------------
// map2token gather FUSED with depthwise 3x3 conv (+bias); fmap stays in L2.
// One block (256 thr = channels) per original point; weighted atomic scatter
// into tok_feat.
// ---------------------------------------------------------------------------
__global__ void map2token_dw(const float* __restrict__ fmap, const float* __restrict__ loc,
                             const int* __restrict__ idx_agg, const float* __restrict__ aggw,
                             const float* __restrict__ dww, const float* __restrict__ dwb,
                             const float* __restrict__ wsum, float* __restrict__ tok_feat) {
    int p = blockIdx.x;
    int b = p >> 14;
    int px, py; point_pixel(loc, p, px, py);
    int t = b * NN + idx_agg[p];
    float w = aggw[p] / (wsum[t] + 1e-6f);
    int c = threadIdx.x;

    float acc = dwb[c];
#pragma unroll
    for (int dy = -1; dy <= 1; ++dy) {
        int yy = py + dy;
        if (yy < 0 || yy >= HH) continue;
#pragma unroll
        for (int dx = -1; dx <= 1; ++dx) {
            int xx = px + dx;
            if (xx < 0 || xx >= WW) continue;
            acc += fmap[(((size_t)b * HH + yy) * WW + xx) * CHID + c]
                 * dww[c * 9 + (dy + 1) * 3 + (dx + 1)];
        }
    }
    atomAddF(&tok_feat[(size_t)t * CHID + c], w * acc);
}

// ---------------------------------------------------------------------------
// skip combine (tok_feat += h*skip_w, in place) + BN2 partial stats
// ---------------------------------------------------------------------------
__global__ void skip_combine_stats(const float* __restrict__ h, float* __restrict__ tok_feat,
                                   const float* __restrict__ skip_w,
                                   float* __restrict__ sum, float* __restrict__ sumsq,
                                   int rows) {
    int c = threadIdx.x;   // 256
    float sw = skip_w[c];
    float s = 0.0f, s2 = 0.0f;
    for (int r = blockIdx.x; r < rows; r += gridDim.x) {
        size_t i = (size_t)r * CHID + c;
        float v = tok_feat[i] + h[i] * sw;
        tok_feat[i] = v;
        s += v; s2 += v * v;
    }
    atomAddF(&sum[c], s);
    atomAddF(&sumsq[c], s2);
}

// ---------------------------------------------------------------------------
// Host-side orchestration
// ---------------------------------------------------------------------------
extern "C" void kernel_launch(void* const* d_in, const int* in_sizes, int n_in,
                              void* d_out, int out_size, void* d_ws, size_t ws_size,
                              hipStream_t stream) {
    const float* x      = (const float*)d_in[0];
    const float* loc    = (const float*)d_in[1];
    const int*   idx    = (const int*)  d_in[2];
    const float* aggw   = (const float*)d_in[3];
    const float* fc1_w  = (const float*)d_in[4];
    const float* fc1_b  = (const float*)d_in[5];
    const float* dw_w   = (const float*)d_in[6];
    const float* dw_b   = (const float*)d_in[7];
    const float* fc2_w  = (const float*)d_in[8];
    const float* fc2_b  = (const float*)d_in[9];
    const float* skip_w = (const float*)d_in[10];
    const float* g1     = (const float*)d_in[11];
    const float* b1     = (const float*)d_in[12];
    const float* g2     = (const float*)d_in[13];
    const float* b2     = (const float*)d_in[14];
    const float* g3     = (const float*)d_in[15];
    const float* b3     = (const float*)d_in[16];
    float* out = (float*)d_out;

    const int ROWS = BB * NN;                         // 32768
    const size_t SZ_H    = (size_t)ROWS * CHID;       // 8,388,608
    const size_t SZ_OUTP = (size_t)ROWS * COUT;       // 2,097,152
    const size_t SZ_FMAP = (size_t)BB * HH * WW * CHID; // 33,554,432
    const size_t SZ_CNT  = (size_t)BB * HH * WW;      // 131,072
    const size_t SZ_WSUM = (size_t)ROWS;              // 32,768

    float* ws      = (float*)d_ws;
    float* h       = ws;
    float* out_pre = h + SZ_H;
    float* fmap    = out_pre + SZ_OUTP;
    float* tokf    = fmap + SZ_FMAP;
    float* counts  = tokf + SZ_H;
    float* wsum    = counts + SZ_CNT;
    float* st      = wsum + SZ_WSUM;
    float* sum1 = st;         float* sq1 = st + 256;
    float* sum2 = st + 512;   float* sq2 = st + 768;
    float* sum3 = st + 1024;  float* sq3 = st + 1088;
    float* sc1  = st + 1152;  float* sh1 = st + 1408;
    float* sc2  = st + 1664;  float* sh2 = st + 1920;
    float* sc3  = st + 2176;  float* sh3 = st + 2240;

    // 1) zero fmap / tok_feat / counts / wsum / stats region (fresh each call)
    size_t zero_n = SZ_FMAP + SZ_H + SZ_CNT + SZ_WSUM + 2304;
    zero_f32<<<4096, 256, 0, stream>>>(fmap, zero_n);

    // 2) fc1:  h[32768,256] = x[32768,64] * fc1_w[256,64]^T + fc1_b
    //    waves = 2048*16 = 32768 -> 4096 blocks of 8 waves
    wmma_gemm_bias<CIN><<<4096, 256, 0, stream>>>(x, fc1_w, fc1_b, h, ROWS, CHID);

    // 3-5) BN1 -> GELU (in place)
    bn_stats<<<1024, CHID, 0, stream>>>(h, sum1, sq1, ROWS);
    bn_finalize<<<1, CHID, 0, stream>>>(sum1, sq1, g1, b1, sc1, sh1, 1.0f / (float)ROWS);
    bn_gelu_apply<<<4096, 256, 0, stream>>>(h, h, sc1, sh1, SZ_H, CHID - 1);

    // 6) token2map scatter-sum (+counts, +wsum)
    token2map<<<BB * N0, CHID, 0, stream>>>(h, loc, idx, aggw, fmap, counts, wsum);

    // 7) divide by counts (mean)
    map_norm<<<4096, 256, 0, stream>>>(fmap, counts, SZ_FMAP);

    // 8) fused depthwise 3x3 + weighted map2token scatter (fmap L2-resident)
    map2token_dw<<<BB * N0, CHID, 0, stream>>>(fmap, loc, idx, aggw, dw_w, dw_b, wsum, tokf);

    // 9-11) skip add + BN2 -> GELU (in place on tokf)
    skip_combine_stats<<<1024, CHID, 0, stream>>>(h, tokf, skip_w, sum2, sq2, ROWS);
    bn_finalize<<<1, CHID, 0, stream>>>(sum2, sq2, g2, b2, sc2, sh2, 1.0f / (float)ROWS);
    bn_gelu_apply<<<4096, 256, 0, stream>>>(tokf, tokf, sc2, sh2, SZ_H, CHID - 1);

    // 12) fc2: out_pre[32768,64] = tokf[32768,256] * fc2_w[64,256]^T + fc2_b
    //     waves = 2048*4 = 8192 -> 1024 blocks
    wmma_gemm_bias<CHID><<<1024, 256, 0, stream>>>(tokf, fc2_w, fc2_b, out_pre, ROWS, COUT);

    // 13-15) BN3 -> GELU -> d_out
    bn_stats<<<1024, COUT, 0, stream>>>(out_pre, sum3, sq3, ROWS);
    bn_finalize<<<1, COUT, 0, stream>>>(sum3, sq3, g3, b3, sc3, sh3, 1.0f / (float)ROWS);
    bn_gelu_apply<<<2048, 256, 0, stream>>>(out_pre, out, sc3, sh3, SZ_OUTP, COUT - 1);
}